// hnode_prompt_layer_feature_sum_21534966022318
// MI455X (gfx1250) — compile-verified
//
#include <hip/hip_runtime.h>

#define BLOCK 256
#define WAVES_PER_BLOCK (BLOCK / 32)

// addrspace(3) pointer type used to recover the 32-bit LDS byte offset
typedef __attribute__((address_space(3))) float lds_float_t;

// ---------------------------------------------------------------------------
// Kernel 1: zero the [N,65] output (harness poisons d_out with 0xAA).
// ---------------------------------------------------------------------------
__global__ void zero_out_kernel(float* __restrict__ out, int n) {
  int i = blockIdx.x * blockDim.x + threadIdx.x;
  int n4 = n >> 2;
  if (i < n4) {
    float4 z = {0.f, 0.f, 0.f, 0.f};
    reinterpret_cast<float4*>(out)[i] = z;
  }
  int tail = n & 3;
  if (i < tail) out[(n4 << 2) + i] = 0.f;
}

// ---------------------------------------------------------------------------
// Kernel 2: gather source rows via async-to-LDS (double buffered per wave),
// then scatter with global f32 atomics (L2-resident output).
//
// Lane layout (wave32): lane = sub*16 + chunk
//   sub   in {0,1}  : which of the 2 edges this wave handles per iteration
//   chunk in [0,16) : which 16-byte (float4) slice of the 256-byte row
// One async instruction per wave moves 2 full rows (512 B) into LDS.
// ---------------------------------------------------------------------------
__global__ void __launch_bounds__(BLOCK)
edge_scatter_kernel(const float* __restrict__ emb,   // [N,64]
                    const float* __restrict__ ef,    // [E]
                    const int*   __restrict__ src,   // [E]
                    const int*   __restrict__ dst,   // [E]
                    float*       __restrict__ out,   // [N,65]
                    int E) {
  // per wave: 2 buffers * 2 rows * 64 floats = 256 floats (1 KB); 8 waves -> 8 KB
  __shared__ float smem[WAVES_PER_BLOCK * 256];

  const int lane  = threadIdx.x & 31;
  const int sub   = lane >> 4;       // 0 or 1
  const int chunk = lane & 15;       // 16B slice index
  const int wid   = threadIdx.x >> 5;
  const int gw    = blockIdx.x * WAVES_PER_BLOCK + wid;
  const int GW    = gridDim.x * WAVES_PER_BLOCK;
  const int P     = (E + 1) >> 1;    // edge pairs

  // 32-bit LDS byte offset of this lane's slot (buffer 0)
  const unsigned lds_base =
      (unsigned)(unsigned long long)(lds_float_t*)(&smem[wid * 256 + lane * 4]);
  const unsigned off0 = lds_base;
  const unsigned off1 = lds_base + 512u;  // second buffer, +128 floats

  // Issue one async gather: each lane pulls 16 B of its edge's source row.
  auto issue = [&](int p, unsigned lds_off) {
    int e  = p * 2 + sub;
    int ec = (e < E) ? e : (E - 1);       // clamp tail lanes in-bounds
    unsigned s = (unsigned)src[ec];
    const float* g = emb + (size_t)s * 64u + (unsigned)(chunk * 4);
    asm volatile("global_load_async_to_lds_b128 %0, %1, off"
                 :: "v"(lds_off), "v"(g)
                 : "memory");
  };

  int p = gw;
  if (p < P) issue(p, off0);              // prime buffer 0
  int buf = 0;
  for (; p < P; p += GW, buf ^= 1) {
    int pn = p + GW;
    if (pn < P) {
      issue(pn, buf ? off0 : off1);       // prefetch next pair into other buffer
      // async "done" returns in order: <=1 outstanding => current buffer ready
      asm volatile("s_wait_asynccnt 0x1" ::: "memory");
    } else {
      asm volatile("s_wait_asynccnt 0x0" ::: "memory");
    }

    int e = p * 2 + sub;
    if (e < E) {
      const float4 v = *reinterpret_cast<const float4*>(
          &smem[wid * 256 + buf * 128 + lane * 4]);   // ds_load_b128
      unsigned d = (unsigned)dst[e];
      float* o = out + (size_t)d * 65u + (unsigned)(chunk * 4);
      atomicAdd(o + 0, v.x);
      atomicAdd(o + 1, v.y);
      atomicAdd(o + 2, v.z);
      atomicAdd(o + 3, v.w);
      if (chunk == 0) {
        atomicAdd(out + (size_t)d * 65u + 64u, ef[e]);  // edge-feature column
      }
    }
  }
}

// ---------------------------------------------------------------------------
extern "C" void kernel_launch(void* const* d_in, const int* in_sizes, int n_in,
                              void* d_out, int out_size, void* d_ws, size_t ws_size,
                              hipStream_t stream) {
  const float* emb = (const float*)d_in[0];   // graph_embedding [N,64] f32
  const float* ef  = (const float*)d_in[1];   // e_feat [E] f32
  const int*   src = (const int*)d_in[2];     // src_idx [E] i32
  const int*   dst = (const int*)d_in[3];     // dst_idx [E] i32
  float* out = (float*)d_out;                 // [N,65] f32
  const int E = in_sizes[1];

  // 1) zero the output
  int n = out_size;
  int zgrid = ((n >> 2) + BLOCK - 1) / BLOCK;
  if (zgrid < 1) zgrid = 1;
  zero_out_kernel<<<zgrid, BLOCK, 0, stream>>>(out, n);

  // 2) gather + scatter-add
  int P = (E + 1) >> 1;
  int blocks = (P + WAVES_PER_BLOCK - 1) / WAVES_PER_BLOCK;
  if (blocks > 1200) blocks = 1200;           // ~60 pairs/wave for pipelining
  if (blocks < 1) blocks = 1;
  edge_scatter_kernel<<<blocks, BLOCK, 0, stream>>>(emb, ef, src, dst, out, E);
}